// NGCF_BPR_55070070670115
// MI455X (gfx1250) — compile-verified
//
#include <hip/hip_runtime.h>
#include <hip/hip_bf16.h>
#include <math.h>

#define N_USERS 100000
#define N_ITEMS 50000
#define N_NODES (N_USERS + N_ITEMS)   // 150000
#define EMB 64
#define NLAYERS 3
#define NNZ 2000000
#define BATCH 4096
#define NEG_SLOPE 0.01f
#define EPS 1e-12f

typedef float v2f __attribute__((ext_vector_type(2)));
typedef float v8f __attribute__((ext_vector_type(8)));

// ---------------------------------------------------------------------------
// Kernel 1: build ego = concat(user_emb, item_emb) and pred[:, 0:64] = ego
// ---------------------------------------------------------------------------
__global__ __launch_bounds__(256) void ngcf_init(
    const float* __restrict__ user_emb, const float* __restrict__ item_emb,
    float* __restrict__ ego, float* __restrict__ pred)
{
    const long long i = (long long)blockIdx.x * 256 + threadIdx.x;
    if (i >= (long long)N_NODES * EMB) return;
    const int row = (int)(i >> 6);
    const int col = (int)(i & 63);
    const float v = (row < N_USERS) ? user_emb[i]
                                    : item_emb[i - (long long)N_USERS * EMB];
    ego[i] = v;
    pred[(size_t)row * 256 + col] = v;
}

// ---------------------------------------------------------------------------
// Kernel 2: COO SpMM scatter:  agg[rows[e]] += vals[e] * ego[cols[e]]
// One wave per edge: 32 lanes x 2 floats = 64-dim row, coalesced row load.
// Scatter uses the hardware GLOBAL_ATOMIC_ADD_F32 (no CAS loop) at device
// scope, since multiple workgroups accumulate into the same L2-resident rows
// (agg = 38.4 MB << 192 MB L2).
// ---------------------------------------------------------------------------
__device__ __forceinline__ void atomic_add_f32_dev(float* p, float x)
{
    asm volatile("global_atomic_add_f32 %0, %1, off scope:SCOPE_DEV"
                 :
                 : "v"(p), "v"(x)
                 : "memory");
}

__global__ __launch_bounds__(256) void ngcf_spmm(
    const float* __restrict__ vals, const int* __restrict__ rows,
    const int* __restrict__ cols, const float* __restrict__ ego,
    float* __restrict__ agg)
{
    const int gtid = blockIdx.x * 256 + threadIdx.x;
    const int e    = gtid >> 5;          // edge index (wave-uniform)
    const int lane = threadIdx.x & 31;
    if (e >= NNZ) return;
    const float v = vals[e];
    const int   r = rows[e];
    const int   c = cols[e];
    const float2 m = *(const float2*)(ego + (size_t)c * EMB + lane * 2);
    float* dst = agg + (size_t)r * EMB + lane * 2;
    atomic_add_f32_dev(dst + 0, v * m.x);
    atomic_add_f32_dev(dst + 1, v * m.y);
}

// ---------------------------------------------------------------------------
// Kernel 3 (fused, WMMA): per 16-row tile
//   aff = ego .* agg
//   h   = leakyrelu(agg @ W1 + aff @ W2 + b1 + b2)    (V_WMMA_F32_16X16X4_F32)
//   aggh <- h (in place; becomes next layer's ego)
//   pred[:, cbase:cbase+64] = h / max(||h||_2, eps)
// Block = 128 threads = 4 waves; each wave computes a 16x64 tile.
//
// f32 WMMA 16x16x4 fragment layouts (ISA 7.12.2):
//   A (16x4):  VGPR0 = K {0 | 2}, VGPR1 = K {1 | 3}; lanes 0-15 / 16-31 = M
//   B (4x16):  VGPR0 = row K {0 | 2}, VGPR1 = row K {1 | 3}; N = lane&15
//   C/D (16x16): VGPR r -> M = r + 8*(lane>>4), N = lane&15
//
// W1/W2 are staged into LDS PRE-SWIZZLED in fragment order so every B operand
// is one aligned ds_load_b64 into a WMMA-ready even VGPR pair (no shuffles):
//   Wp[frag=kk*4+nn][lane] = { W[kb][nb], W[kb+1][nb] },
//   kb = kk*4 + (lane>>4)*2,  nb = nn*16 + (lane&15).
// 32 lanes x 8 B = 256 B = one full 64-bank sweep -> conflict-free.
// ---------------------------------------------------------------------------
__global__ __launch_bounds__(128) void ngcf_layer_wmma(
    const float* __restrict__ ego, float* __restrict__ aggh,
    const float* __restrict__ W1, const float* __restrict__ b1,
    const float* __restrict__ W2, const float* __restrict__ b2,
    float* __restrict__ pred, int cbase)
{
    __shared__ float Wp1[64 * 64];      // 64 fragments x 32 lanes x 2 floats
    __shared__ float Wp2[64 * 64];
    __shared__ float bsum[64];
    __shared__ float hs[4 * 16 * 64];   // 4 KB per-wave h scratch

    const int tid = threadIdx.x;
    for (int idx = tid; idx < 64 * 32; idx += 128) {
        const int frag = idx >> 5;           // kk*4 + nn
        const int ln   = idx & 31;
        const int kk = frag >> 2, nn = frag & 3;
        const int kb = kk * 4 + (ln >> 4) * 2;
        const int nb = nn * 16 + (ln & 15);
        float2 w1, w2;
        w1.x = W1[(kb + 0) * 64 + nb];
        w1.y = W1[(kb + 1) * 64 + nb];
        w2.x = W2[(kb + 0) * 64 + nb];
        w2.y = W2[(kb + 1) * 64 + nb];
        *(float2*)(Wp1 + idx * 2) = w1;
        *(float2*)(Wp2 + idx * 2) = w2;
    }
    if (tid < 64) bsum[tid] = b1[tid] + b2[tid];
    __syncthreads();

    const int wave = tid >> 5;
    const int lane = tid & 31;
    const int half = lane >> 4;          // 0 or 1
    const int l15  = lane & 15;
    const int rowBase = (blockIdx.x * 4 + wave) * 16;
    if (rowBase >= N_NODES) return;      // wave-uniform; no barriers below

    // ---- A fragments for all 16 K-steps (agg and aff = ego .* agg) ----
    const float* egoRow = ego  + (size_t)(rowBase + l15) * EMB;
    const float* aggRow = aggh + (size_t)(rowBase + l15) * EMB;
    v2f Aagg[16], Aaff[16];
#pragma unroll
    for (int kk = 0; kk < 16; ++kk) {
        const int kc = kk * 4 + half * 2;
        const float2 g = *(const float2*)(aggRow + kc);
        const float2 e = *(const float2*)(egoRow + kc);
        v2f a; a[0] = g.x;       a[1] = g.y;       Aagg[kk] = a;
        v2f f; f[0] = g.x * e.x; f[1] = g.y * e.y; Aaff[kk] = f;
    }

    // ---- accumulators initialized with bias (same col for all 8 rows) ----
    v8f acc[4];
#pragma unroll
    for (int nn = 0; nn < 4; ++nn) {
        const float b = bsum[nn * 16 + l15];
#pragma unroll
        for (int t = 0; t < 8; ++t) acc[nn][t] = b;
    }

    // ---- main WMMA loop: 16 K-steps x 4 N-tiles x 2 matrices ----
    const float* fb1 = Wp1 + lane * 2;
    const float* fb2 = Wp2 + lane * 2;
#pragma unroll
    for (int kk = 0; kk < 16; ++kk) {
#pragma unroll
        for (int nn = 0; nn < 4; ++nn) {
            const int foff = (kk * 4 + nn) * 64;   // fragment stride: 32 lanes * 2 f
            const v2f B1 = *(const v2f*)(fb1 + foff);
            const v2f B2 = *(const v2f*)(fb2 + foff);
            acc[nn] = __builtin_amdgcn_wmma_f32_16x16x4_f32(
                false, Aagg[kk], false, B1, (short)0, acc[nn], false, false);
            acc[nn] = __builtin_amdgcn_wmma_f32_16x16x4_f32(
                false, Aaff[kk], false, B2, (short)0, acc[nn], false, false);
        }
    }

    // ---- leaky relu; write h in place (next ego) + to LDS scratch ----
    float* hsw = hs + wave * (16 * 64);
#pragma unroll
    for (int nn = 0; nn < 4; ++nn) {
#pragma unroll
        for (int r = 0; r < 8; ++r) {
            float x = acc[nn][r];
            x = (x > 0.0f) ? x : NEG_SLOPE * x;
            const int M   = r + half * 8;
            const int col = nn * 16 + l15;
            aggh[(size_t)(rowBase + M) * EMB + col] = x;
            hsw[M * 64 + col] = x;
        }
    }

    // ---- row-wise L2 norm -> pred columns (lanes 0..15, one row each) ----
    // Same-wave DS ops are processed in order, so hsw reads see the writes.
    if (half == 0) {
        const float* hrow = hsw + l15 * 64;
        float ss = 0.0f;
#pragma unroll
        for (int c = 0; c < 64; c += 4) {
            const float4 t = *(const float4*)(hrow + c);
            ss += t.x * t.x + t.y * t.y + t.z * t.z + t.w * t.w;
        }
        const float scale = 1.0f / fmaxf(sqrtf(ss), EPS);
        float* prow = pred + (size_t)(rowBase + l15) * 256 + cbase;
#pragma unroll
        for (int c = 0; c < 64; c += 4) {
            float4 t = *(const float4*)(hrow + c);
            t.x *= scale; t.y *= scale; t.z *= scale; t.w *= scale;
            *(float4*)(prow + c) = t;
        }
    }
}

// ---------------------------------------------------------------------------
// Kernel 4: gather 3 x 4096 rows of pred into d_out (BPR triplets)
// ---------------------------------------------------------------------------
__global__ __launch_bounds__(64) void ngcf_gather(
    const float* __restrict__ pred, const int* __restrict__ u,
    const int* __restrict__ ii, const int* __restrict__ jj,
    float* __restrict__ out)
{
    const int b   = blockIdx.x;
    const int sel = b >> 12;         // BATCH = 4096 = 2^12
    const int s   = b & (BATCH - 1);
    int row;
    if (sel == 0)      row = u[s];
    else if (sel == 1) row = N_USERS + ii[s];
    else               row = N_USERS + jj[s];
    const float4* src = (const float4*)(pred + (size_t)row * 256);
    float4* dst = (float4*)(out + (size_t)b * 256);
    dst[threadIdx.x] = src[threadIdx.x];   // 64 lanes x float4 = 256 floats
}

// ---------------------------------------------------------------------------
// Host launcher.
// Workspace layout (floats):
//   [0, 9.6M)        bufA : ego ping     (N_NODES*64)
//   [9.6M, 19.2M)    bufB : agg/h pong   (N_NODES*64)
//   [19.2M, 57.6M)   pred : N_NODES*256
// total = 230.4 MB
// ---------------------------------------------------------------------------
extern "C" void kernel_launch(void* const* d_in, const int* in_sizes, int n_in,
                              void* d_out, int out_size, void* d_ws, size_t ws_size,
                              hipStream_t stream)
{
    const float* user_emb = (const float*)d_in[0];
    const float* item_emb = (const float*)d_in[1];
    const float* W1w      = (const float*)d_in[2];
    const float* W1b      = (const float*)d_in[3];
    const float* W2w      = (const float*)d_in[4];
    const float* W2b      = (const float*)d_in[5];
    const float* vals     = (const float*)d_in[6];
    const int*   rows     = (const int*)d_in[7];
    const int*   cols     = (const int*)d_in[8];
    const int*   u_idx    = (const int*)d_in[9];
    const int*   i_idx    = (const int*)d_in[10];
    const int*   j_idx    = (const int*)d_in[11];

    float* ws = (float*)d_ws;
    const size_t egoElems = (size_t)N_NODES * EMB;
    float* bufA = ws;
    float* bufB = ws + egoElems;
    float* pred = ws + 2 * egoElems;

    {
        const long long total = (long long)N_NODES * EMB;
        const int grid = (int)((total + 255) / 256);
        ngcf_init<<<grid, 256, 0, stream>>>(user_emb, item_emb, bufA, pred);
    }

    float* cur = bufA;   // ego input
    float* nxt = bufB;   // agg -> h (in place)
    const int spmmGrid  = (int)(((long long)NNZ * 32 + 255) / 256);
    const int tiles     = N_NODES / 16;                 // 9375
    const int fusedGrid = (tiles + 3) / 4;              // 2344

    for (int k = 0; k < NLAYERS; ++k) {
        hipMemsetAsync(nxt, 0, egoElems * sizeof(float), stream);
        ngcf_spmm<<<spmmGrid, 256, 0, stream>>>(vals, rows, cols, cur, nxt);
        ngcf_layer_wmma<<<fusedGrid, 128, 0, stream>>>(
            cur, nxt,
            W1w + (size_t)k * 64 * 64, W1b + (size_t)k * 64,
            W2w + (size_t)k * 64 * 64, W2b + (size_t)k * 64,
            pred, 64 * (k + 1));
        float* t = cur; cur = nxt; nxt = t;   // h becomes next ego
    }

    ngcf_gather<<<3 * BATCH, 64, 0, stream>>>(pred, u_idx, i_idx, j_idx,
                                              (float*)d_out);
}